// ContrastStretch_34059090657778
// MI455X (gfx1250) — compile-verified
//
#include <hip/hip_runtime.h>
#include <hip/hip_bf16.h>
#include <cstdint>
#include <cmath>

// ---------------- constants ----------------
#define TPB      256           // threads per block (8 wave32 waves)
#define CHUNK    2048          // floats per staged tile (8 KB) -> double buffered
#define IPW      2             // async-load instructions per wave per full tile:
                               // CHUNK/4 float4 loads / 256 threads = 2 per thread,
                               // and each unrolled asm stmt issues once per wave
#define NB0      2048          // level-0 bins: key[31:21]
#define NB1      2048          // level-1 bins: key[20:10]
#define NB2      1024          // level-2 bins: key[9:0]
#define NBATCH   64
#define BLK_PER_BATCH 32       // hist blocks per batch row (each loops over chunks)

// ---------------- gfx1250 async global->LDS helpers ----------------
__device__ __forceinline__ void async_b128_to_lds(uint32_t lds_byte_off, const void* gptr) {
    // GLOBAL_LOAD_ASYNC_TO_LDS_B128 (GV mode): vdst = LDS byte address VGPR,
    // vaddr = 64-bit global address VGPR pair. Tracked by ASYNCcnt.
    asm volatile("global_load_async_to_lds_b128 %0, %1, off"
                 :: "v"(lds_byte_off), "v"((uint64_t)(uintptr_t)gptr)
                 : "memory");
}
template <int NCNT>
__device__ __forceinline__ void wait_async_le() {
    // Partial wait: ASYNCcnt <= NCNT. Async loads complete in order, so waiting
    // down to the younger tile's instruction count proves the older tile landed.
    asm volatile("s_wait_asynccnt %0" :: "i"(NCNT) : "memory");
}
// Generic pointer to LDS carries the LDS byte offset in its low 32 bits.
__device__ __forceinline__ uint32_t lds_off(const void* p) {
    return (uint32_t)(uintptr_t)p;
}

// ---------------- order-preserving float <-> uint key ----------------
__device__ __forceinline__ uint32_t f2key(float f) {
    uint32_t u = __float_as_uint(f);
    return u ^ ((u >> 31) ? 0xFFFFFFFFu : 0x80000000u);
}
__device__ __forceinline__ float key2f(uint32_t k) {
    uint32_t u = k ^ ((k >> 31) ? 0x80000000u : 0xFFFFFFFFu);
    return __uint_as_float(u);
}

// Issue staging of one tile. Full tiles use the async engine (wave-uniform,
// exactly IPW instructions per wave); partial tails use scalar staging so the
// per-wave ASYNCcnt bookkeeping stays exact. Returns true if async was used.
__device__ __forceinline__ bool issue_stage(float* buf, const float* gsrc, int remaining) {
    if (remaining >= CHUNK) {
        const uint32_t b = lds_off(buf);
#pragma unroll
        for (int j = 0; j < CHUNK / 4 / TPB; ++j) {
            int off = (j * TPB + (int)threadIdx.x) * 4;   // float index in tile
            async_b128_to_lds(b + (uint32_t)off * 4u, gsrc + off);
        }
        return true;
    }
    for (int i = threadIdx.x; i < remaining; i += TPB) buf[i] = gsrc[i];
    return false;
}

// ---------------- generic double-buffered histogram driver ----------------
// process(elem_key) is inlined per kernel via a macro to keep things simple.
#define HIST_PIPELINE(PROCESS_BODY)                                             \
    const int batch    = blockIdx.y;                                            \
    const long long gb = (long long)batch * N;                                  \
    const int nChunks  = (N + CHUNK - 1) / CHUNK;                               \
    const int stride   = gridDim.x;                                             \
    int par = 0;                                                                \
    int c0  = blockIdx.x;                                                       \
    if (c0 < nChunks)                                                           \
        issue_stage(tiles[0], x + gb + (long long)c0 * CHUNK, N - c0 * CHUNK);  \
    for (int c = c0; c < nChunks; c += stride) {                                \
        int cn = c + stride;                                                    \
        bool nextAsync = false;                                                 \
        if (cn < nChunks)                                                       \
            nextAsync = issue_stage(tiles[par ^ 1],                             \
                                    x + gb + (long long)cn * CHUNK,             \
                                    N - cn * CHUNK);                            \
        if (nextAsync) wait_async_le<IPW>(); else wait_async_le<0>();           \
        __syncthreads();                                                        \
        {                                                                       \
            const float* tile = tiles[par];                                     \
            const int rem = N - c * CHUNK;                                      \
            const int cnt = rem < CHUNK ? rem : CHUNK;                          \
            _Pragma("unroll")                                                   \
            for (int j = 0; j < CHUNK / TPB; ++j) {                             \
                int idx = j * TPB + (int)threadIdx.x;                           \
                if (idx < cnt) {                                                \
                    uint32_t k = f2key(tile[idx]);                              \
                    PROCESS_BODY                                                \
                }                                                               \
            }                                                                   \
        }                                                                       \
        __syncthreads();                                                        \
        par ^= 1;                                                               \
    }

// ---------------- level 0 histogram ----------------
__global__ void hist0_kernel(const float* __restrict__ x, uint32_t* __restrict__ H0, int N) {
    __shared__ float    tiles[2][CHUNK];     // 16 KB
    __shared__ uint32_t hist[NB0];           //  8 KB
    for (int i = threadIdx.x; i < NB0; i += TPB) hist[i] = 0;
    __syncthreads();

    HIST_PIPELINE( atomicAdd(&hist[k >> 21], 1u); )

    __syncthreads();
    uint32_t* g = H0 + (size_t)blockIdx.y * NB0;
    for (int i = threadIdx.x; i < NB0; i += TPB) {
        uint32_t c = hist[i];
        if (c) atomicAdd(&g[i], c);
    }
}

// ---------------- level 1 histogram (filtered on level-0 prefix) ----------------
__global__ void hist1_kernel(const float* __restrict__ x, uint32_t* __restrict__ H1,
                             const uint32_t* __restrict__ prefA, int N) {
    __shared__ float    tiles[2][CHUNK];     // 16 KB
    __shared__ uint32_t hist[4 * NB1];       // 32 KB
    const uint32_t p0 = prefA[blockIdx.y * 4 + 0], p1 = prefA[blockIdx.y * 4 + 1];
    const uint32_t p2 = prefA[blockIdx.y * 4 + 2], p3 = prefA[blockIdx.y * 4 + 3];
    for (int i = threadIdx.x; i < 4 * NB1; i += TPB) hist[i] = 0;
    __syncthreads();

    HIST_PIPELINE(
        uint32_t d0 = k >> 21;
        uint32_t d1 = (k >> 10) & (NB1 - 1);
        if (d0 == p0) atomicAdd(&hist[0 * NB1 + d1], 1u);
        if (d0 == p1) atomicAdd(&hist[1 * NB1 + d1], 1u);
        if (d0 == p2) atomicAdd(&hist[2 * NB1 + d1], 1u);
        if (d0 == p3) atomicAdd(&hist[3 * NB1 + d1], 1u);
    )

    __syncthreads();
    uint32_t* g = H1 + (size_t)blockIdx.y * 4 * NB1;
    for (int i = threadIdx.x; i < 4 * NB1; i += TPB) {
        uint32_t c = hist[i];
        if (c) atomicAdd(&g[i], c);
    }
}

// ---------------- level 2 histogram (filtered on 22-bit prefix) ----------------
__global__ void hist2_kernel(const float* __restrict__ x, uint32_t* __restrict__ H2,
                             const uint32_t* __restrict__ prefA,
                             const uint32_t* __restrict__ prefB, int N) {
    __shared__ float    tiles[2][CHUNK];     // 16 KB
    __shared__ uint32_t hist[4 * NB2];       // 16 KB
    uint32_t q[4];
#pragma unroll
    for (int t = 0; t < 4; ++t)
        q[t] = (prefA[blockIdx.y * 4 + t] << 11) | prefB[blockIdx.y * 4 + t];
    for (int i = threadIdx.x; i < 4 * NB2; i += TPB) hist[i] = 0;
    __syncthreads();

    HIST_PIPELINE(
        uint32_t top = k >> 10;
        uint32_t d2  = k & (NB2 - 1);
        if (top == q[0]) atomicAdd(&hist[0 * NB2 + d2], 1u);
        if (top == q[1]) atomicAdd(&hist[1 * NB2 + d2], 1u);
        if (top == q[2]) atomicAdd(&hist[2 * NB2 + d2], 1u);
        if (top == q[3]) atomicAdd(&hist[3 * NB2 + d2], 1u);
    )

    __syncthreads();
    uint32_t* g = H2 + (size_t)blockIdx.y * 4 * NB2;
    for (int i = threadIdx.x; i < 4 * NB2; i += TPB) {
        uint32_t c = hist[i];
        if (c) atomicAdd(&g[i], c);
    }
}

// ---------------- selection scans ----------------
__global__ void select0_kernel(const uint32_t* __restrict__ H0,
                               uint32_t r0, uint32_t r1, uint32_t r2, uint32_t r3,
                               uint32_t* __restrict__ prefA, uint32_t* __restrict__ rankA) {
    int i = threadIdx.x;                 // one block, NBATCH*4 threads
    if (i >= NBATCH * 4) return;
    int b = i >> 2, t = i & 3;
    uint32_t r = (t == 0) ? r0 : (t == 1) ? r1 : (t == 2) ? r2 : r3;
    const uint32_t* h = H0 + (size_t)b * NB0;
    uint32_t cum = 0, bin = 0;
    for (; bin < NB0 - 1; ++bin) {
        uint32_t c = h[bin];
        if (cum + c > r) break;
        cum += c;
    }
    prefA[i] = bin;
    rankA[i] = r - cum;
}

__global__ void select1_kernel(const uint32_t* __restrict__ H1,
                               const uint32_t* __restrict__ rankA,
                               uint32_t* __restrict__ prefB, uint32_t* __restrict__ rankB) {
    int i = threadIdx.x;
    if (i >= NBATCH * 4) return;
    int b = i >> 2, t = i & 3;
    uint32_t r = rankA[i];
    const uint32_t* h = H1 + ((size_t)b * 4 + t) * NB1;
    uint32_t cum = 0, bin = 0;
    for (; bin < NB1 - 1; ++bin) {
        uint32_t c = h[bin];
        if (cum + c > r) break;
        cum += c;
    }
    prefB[i] = bin;
    rankB[i] = r - cum;
}

__global__ void select2_kernel(const uint32_t* __restrict__ H2,
                               const uint32_t* __restrict__ prefA,
                               const uint32_t* __restrict__ prefB,
                               const uint32_t* __restrict__ rankB,
                               float* __restrict__ scaleoff,
                               float fracLo, float fracHi, float eps) {
    __shared__ float vals[NBATCH * 4];
    int i = threadIdx.x;
    if (i < NBATCH * 4) {
        int b = i >> 2, t = i & 3;
        uint32_t r = rankB[i];
        const uint32_t* h = H2 + ((size_t)b * 4 + t) * NB2;
        uint32_t cum = 0, bin = 0;
        for (; bin < NB2 - 1; ++bin) {
            uint32_t c = h[bin];
            if (cum + c > r) break;
            cum += c;
        }
        uint32_t key = (prefA[i] << 21) | (prefB[i] << 10) | bin;
        vals[i] = key2f(key);
    }
    __syncthreads();
    if (i < NBATCH * 4 && (i & 3) == 0) {
        float v0 = vals[i], v1 = vals[i + 1], v2 = vals[i + 2], v3 = vals[i + 3];
        float lo = v0 + fracLo * (v1 - v0);
        float hi = v2 + fracHi * (v3 - v2);
        int b = i >> 2;
        scaleoff[2 * b]     = lo;
        scaleoff[2 * b + 1] = 1.0f / (hi - lo + eps);
    }
}

// ---------------- streaming normalize ----------------
__global__ void normalize_kernel(const float* __restrict__ x, float* __restrict__ y,
                                 const float* __restrict__ so, int N) {
    const int batch = blockIdx.y;
    const long long base = (long long)batch * N;
    const float lo  = so[2 * batch];
    const float inv = so[2 * batch + 1];
    const int n4 = N >> 2;
    const float4* xin = (const float4*)(x + base);
    float4* yout = (float4*)(y + base);
    int i = blockIdx.x * blockDim.x + threadIdx.x;
    if (i < n4) {
        float4 v = xin[i];
        v.x = fminf(fmaxf((v.x - lo) * inv, 0.0f), 1.0f);
        v.y = fminf(fmaxf((v.y - lo) * inv, 0.0f), 1.0f);
        v.z = fminf(fmaxf((v.z - lo) * inv, 0.0f), 1.0f);
        v.w = fminf(fmaxf((v.w - lo) * inv, 0.0f), 1.0f);
        yout[i] = v;
    }
    if (blockIdx.x == 0 && (int)threadIdx.x < (N & 3)) {   // scalar tail
        int idx = (n4 << 2) + (int)threadIdx.x;
        float v = x[base + idx];
        y[base + idx] = fminf(fmaxf((v - lo) * inv, 0.0f), 1.0f);
    }
}

// ---------------- host launcher ----------------
extern "C" void kernel_launch(void* const* d_in, const int* in_sizes, int n_in,
                              void* d_out, int out_size, void* d_ws, size_t ws_size,
                              hipStream_t stream) {
    const float* x = (const float*)d_in[0];
    float* y = (float*)d_out;

    const int Bn = NBATCH;                      // batch dim of reference input
    const int N  = in_sizes[0] / Bn;            // elements per batch row

    // workspace layout (uint32 words)
    uint32_t* H0    = (uint32_t*)d_ws;                           // Bn*NB0
    uint32_t* H1    = H0 + (size_t)Bn * NB0;                     // Bn*4*NB1
    uint32_t* H2    = H1 + (size_t)Bn * 4 * NB1;                 // Bn*4*NB2
    uint32_t* prefA = H2 + (size_t)Bn * 4 * NB2;                 // Bn*4
    uint32_t* rankA = prefA + (size_t)Bn * 4;
    uint32_t* prefB = rankA + (size_t)Bn * 4;
    uint32_t* rankB = prefB + (size_t)Bn * 4;
    float*    so    = (float*)(rankB + (size_t)Bn * 4);          // Bn*2

    const size_t histBytes = ((size_t)Bn * NB0 + (size_t)Bn * 4 * NB1 +
                              (size_t)Bn * 4 * NB2) * sizeof(uint32_t);
    hipMemsetAsync(d_ws, 0, histBytes, stream);

    // quantile rank arithmetic (jnp.quantile linear interpolation)
    const double posLo = 0.05 * (double)(N - 1);
    const double posHi = 0.95 * (double)(N - 1);
    const uint32_t kLo = (uint32_t)posLo;
    const uint32_t kHi = (uint32_t)posHi;
    const float fracLo = (float)(posLo - (double)kLo);
    const float fracHi = (float)(posHi - (double)kHi);
    const uint32_t nMax = (uint32_t)(N - 1);
    const uint32_t r0 = kLo, r1 = (kLo + 1 < nMax) ? kLo + 1 : nMax;
    const uint32_t r2 = kHi, r3 = (kHi + 1 < nMax) ? kHi + 1 : nMax;

    const int nChunks = (N + CHUNK - 1) / CHUNK;
    const int bpb = nChunks < BLK_PER_BATCH ? nChunks : BLK_PER_BATCH;
    dim3 gHist(bpb, Bn);

    hist0_kernel<<<gHist, TPB, 0, stream>>>(x, H0, N);
    select0_kernel<<<1, 256, 0, stream>>>(H0, r0, r1, r2, r3, prefA, rankA);
    hist1_kernel<<<gHist, TPB, 0, stream>>>(x, H1, prefA, N);
    select1_kernel<<<1, 256, 0, stream>>>(H1, rankA, prefB, rankB);
    hist2_kernel<<<gHist, TPB, 0, stream>>>(x, H2, prefA, prefB, N);
    select2_kernel<<<1, 256, 0, stream>>>(H2, prefA, prefB, rankB, so,
                                          fracLo, fracHi, 1e-6f);

    const int n4 = N >> 2;
    dim3 gNorm((n4 + TPB - 1) / TPB, Bn);
    normalize_kernel<<<gNorm, TPB, 0, stream>>>(x, y, so, N);
}